// HungarianMatcher_17274358464826
// MI455X (gfx1250) — compile-verified
//
#include <hip/hip_runtime.h>

// CDNA5 / gfx1250 Hungarian-matcher cost kernel.
// One wave32 per (bs,seq) sample: q=32 queries x V=16 targets cost tile.
// Class cost is computed as a 16x16x4 f32 WMMA against a one-hot matrix
// (K = ncls = 4 exactly matches V_WMMA_F32_16X16X4_F32), with the bbox-L1 and
// GIoU terms pre-accumulated into the WMMA C operand (D = A*B + C fusion).

typedef __attribute__((ext_vector_type(2))) float v2f;
typedef __attribute__((ext_vector_type(8))) float v8f;

#define COST_CLASS_W 1.0f
#define COST_BBOX_W  5.0f
#define COST_GIOU_W  2.0f

__global__ __launch_bounds__(256) void matcher_cost_wmma(
    const float* __restrict__ logits,   // [n,32,4]  n = bs*seq = 256
    const float* __restrict__ pboxes,   // [n,32,4]  cxcywh
    const float* __restrict__ tboxes,   // [n,32,4]  cxcywh (first 16 used)
    const int*   __restrict__ ids,      // [n,32]    (first 16 used)
    float*       __restrict__ out,      // [n,32,16]
    int n)
{
    __shared__ float4 spbox[8][32];     // 8 waves/block, 32 pred boxes each

    const int lane = threadIdx.x & 31;
    const int wid  = threadIdx.x >> 5;
    const int s    = blockIdx.x * 8 + wid;
    if (s >= n) return;

    // Stage this sample's 32 pred boxes in LDS (one float4 per lane).
    spbox[wid][lane] = reinterpret_cast<const float4*>(pboxes)[s * 32 + lane];
    __syncthreads();

    const int j  = lane & 15;   // target column (WMMA N)
    const int hi = lane >> 4;   // half-wave select
    const int k0 = hi * 2;      // K base held by this lane (A/B operand layout)

    // Target box (fixed per lane): cxcywh -> xyxy + area.
    const float4 tb = reinterpret_cast<const float4*>(tboxes)[s * 32 + j];
    const float tx0 = tb.x - 0.5f * tb.z, ty0 = tb.y - 0.5f * tb.w;
    const float tx1 = tb.x + 0.5f * tb.z, ty1 = tb.y + 0.5f * tb.w;
    const float tarea = tb.z * tb.w;

    // One-hot B matrix [4 x 16] with -COST_CLASS folded in:
    // B[k][j] = (ids[j]==k) ? -COST_CLASS : 0. Lane holds K = {k0, k0+1}.
    const int id = ids[s * 32 + j];
    v2f b;
    b.x = (id == k0    ) ? -COST_CLASS_W : 0.0f;
    b.y = (id == k0 + 1) ? -COST_CLASS_W : 0.0f;

    const float2* lg2 = reinterpret_cast<const float2*>(logits);

    #pragma unroll
    for (int tile = 0; tile < 2; ++tile) {
        // A matrix tile [16 x 4]: lane holds (M = tile*16 + lane%16, K = {k0,k0+1}).
        const int m = tile * 16 + (lane & 15);
        const float2 av = lg2[(s * 32 + m) * 2 + hi];
        v2f a; a.x = av.x; a.y = av.y;

        // Accumulate bbox-L1 + GIoU costs in the C/D register layout:
        // VGPR r, lane l -> row i = tile*16 + 8*(l>=16) + r, col j = l%16.
        v8f acc;
        #pragma unroll
        for (int r = 0; r < 8; ++r) {
            const int i = tile * 16 + hi * 8 + r;
            const float4 pb = spbox[wid][i];
            const float px0 = pb.x - 0.5f * pb.z, py0 = pb.y - 0.5f * pb.w;
            const float px1 = pb.x + 0.5f * pb.z, py1 = pb.y + 0.5f * pb.w;
            const float parea = pb.z * pb.w;

            // L1 in cxcywh space (matches reference cdist on raw boxes).
            const float l1 = fabsf(pb.x - tb.x) + fabsf(pb.y - tb.y)
                           + fabsf(pb.z - tb.z) + fabsf(pb.w - tb.w);

            // Intersection / union.
            float iw = fminf(px1, tx1) - fmaxf(px0, tx0); iw = fmaxf(iw, 0.0f);
            float ih = fminf(py1, ty1) - fmaxf(py0, ty0); ih = fmaxf(ih, 0.0f);
            const float inter = iw * ih;
            const float uni   = parea + tarea - inter;
            const float iou   = inter / uni;

            // Smallest enclosing box.
            float ew = fmaxf(px1, tx1) - fminf(px0, tx0); ew = fmaxf(ew, 0.0f);
            float eh = fmaxf(py1, ty1) - fminf(py0, ty0); eh = fmaxf(eh, 0.0f);
            const float earea = ew * eh;
            const float giou  = iou - (earea - uni) / earea;

            acc[r] = COST_BBOX_W * l1 - COST_GIOU_W * giou;
        }

        // D = A x B + acc : fused class-cost matmul + other costs.
        // 8 args: (neg_a, A, neg_b, B, c_mod, C, reuse_a, reuse_b)
        v8f d = __builtin_amdgcn_wmma_f32_16x16x4_f32(
            false, a, false, b, (short)0, acc, false, false);

        // Store: element r -> out[s][tile*16 + hi*8 + r][j].
        // For fixed r, lanes 0-15 / 16-31 each cover one contiguous 64B row.
        float* o = out + s * 512 + tile * 256 + (hi * 8) * 16 + j;
        #pragma unroll
        for (int r = 0; r < 8; ++r)
            o[r * 16] = d[r];
    }
}

extern "C" void kernel_launch(void* const* d_in, const int* in_sizes, int n_in,
                              void* d_out, int out_size, void* d_ws, size_t ws_size,
                              hipStream_t stream) {
    const float* logits = (const float*)d_in[0];   // pred_logits [8,32,32,4]
    const float* pboxes = (const float*)d_in[1];   // pred_boxes  [8,32,32,4]
    const float* tboxes = (const float*)d_in[2];   // tgt_boxes   [8,32,32,4]
    const int*   ids    = (const int*)  d_in[3];   // hand_type   [8,32,32]
    float*       out    = (float*)d_out;           // [256,32,16]

    const int n = in_sizes[3] / 32;                // bs*seq = 256
    const int blocks = (n + 7) / 8;                // 8 waves (samples) per block
    matcher_cost_wmma<<<blocks, 256, 0, stream>>>(logits, pboxes, tboxes, ids, out, n);
}